// ChamferNormalLoss_69346541961758
// MI455X (gfx1250) — compile-verified
//
#include <hip/hip_runtime.h>
#include <hip/hip_bf16.h>

typedef __attribute__((ext_vector_type(2))) float v2f;
typedef __attribute__((ext_vector_type(8))) float v8f;

#define B_   8
#define P_   2048
#define V_   2562
#define N_   8192
#define F_   16384
#define VPAD 2576            // ceil(2562/16)*16
#define TPAD 32              // 2 extra ref tiles of padding for pipelined loads
#define EPSF 1e-12f

// ---------------------------------------------------------------------------
// Build packed ref table: [B, Rpad+TPAD] of float4 { -2x, -2y, -2z, |r|^2 }.
// Padded slots get |r|^2 = 3e38 so they can never win the argmin; the extra
// TPAD entries let the NN kernel software-pipeline loads without bounds checks.
// ---------------------------------------------------------------------------
__global__ void build_reftab(const float* __restrict__ refs,
                             float4* __restrict__ tab,
                             int R, int Rtot) {
    int i = blockIdx.x * blockDim.x + threadIdx.x;
    int total = B_ * Rtot;
    if (i >= total) return;
    int b = i / Rtot;
    int r = i - b * Rtot;
    float4 o;
    if (r < R) {
        const float* p = refs + ((size_t)b * R + r) * 3;
        float x = p[0], y = p[1], z = p[2];
        o = make_float4(-2.0f * x, -2.0f * y, -2.0f * z, x * x + y * y + z * z);
    } else {
        o = make_float4(0.0f, 0.0f, 0.0f, 3.0e38f);
    }
    tab[i] = o;
}

// ---------------------------------------------------------------------------
// D = A x B + 0 with V_WMMA_F32_16X16X4_F32.
// A = [x y z 1] (16x4 queries), B = [-2r | rr] (4x16 refs)  =>  D = rr - 2 q.r
// ---------------------------------------------------------------------------
__device__ __forceinline__ v8f wmma4(v2f a, v2f b) {
    v8f c = {};
    return __builtin_amdgcn_wmma_f32_16x16x4_f32(
        /*neg_a=*/false, a, /*neg_b=*/false, b,
        /*c_mod=*/(short)0, c, /*reuse_a=*/false, /*reuse_b=*/false);
}

#define UPDATE(BV, BI, C, NIDX)                                        \
    _Pragma("unroll")                                                  \
    for (int _i = 0; _i < 8; ++_i) {                                   \
        float _v = (C)[_i];                                            \
        if (_v < (BV)[_i]) { (BV)[_i] = _v; (BI)[_i] = (NIDX); }       \
    }

// Cross-lane reduce the 16 columns in each half-wave (tie -> smaller index)
// and write 16 indices for one query tile.
__device__ __forceinline__ void reduce_write(float* bv, int* bi,
                                             int* __restrict__ outIdx,
                                             int base, int col, int hi) {
#pragma unroll
    for (int i = 0; i < 8; ++i) {
#pragma unroll
        for (int off = 8; off >= 1; off >>= 1) {
            float ov = __shfl_xor(bv[i], off, 16);
            int   oi = __shfl_xor(bi[i], off, 16);
            if (ov < bv[i] || (ov == bv[i] && oi < bi[i])) {
                bv[i] = ov; bi[i] = oi;
            }
        }
        if (col == 0) outIdx[base + i + 8 * hi] = bi[i];
    }
}

// ---------------------------------------------------------------------------
// Nearest-neighbor argmin. One wave handles 32 queries (2 query tiles) and
// sweeps ref tiles two at a time with register double-buffered B loads:
// 4 WMMAs + 2 x b64 loads per iteration, 4 independent min chains.
// ---------------------------------------------------------------------------
__global__ __launch_bounds__(32) void nn_argmin_wmma(
    const float*  __restrict__ pts,   // [B, P, 3]
    const float4* __restrict__ tab,   // [B, Rpad+TPAD] packed
    int*          __restrict__ outIdx,// [B, P]
    int Rpad) {
    const int lane  = threadIdx.x;            // 0..31
    const int blk   = blockIdx.x;             // B * (P/32)
    const int b     = blk / (P_ / 32);
    const int qbase = (blk - b * (P_ / 32)) * 32;
    const int col   = lane & 15;              // N column / M row
    const int hi    = lane >> 4;              // K half selector

    // A operands for the two query tiles:
    // lanes 0-15 hold (K=0,1)=(x,y); lanes 16-31 hold (K=2,3)=(z,1)
    v2f a0, a1;
    {
        const float* q0 = pts + ((size_t)b * P_ + qbase + col) * 3;
        const float* q1 = q0 + 16 * 3;
        a0[0] = (hi == 0) ? q0[0] : q0[2];
        a0[1] = (hi == 0) ? q0[1] : 1.0f;
        a1[0] = (hi == 0) ? q1[0] : q1[2];
        a1[1] = (hi == 0) ? q1[1] : 1.0f;
    }

    // Each lane's B slot for ref n is bt[2*n + hi] (8 bytes).
    const float2* bt = (const float2*)(tab + (size_t)b * (Rpad + TPAD));

    // 4 independent accumulator sets: [query tile][ref-tile parity]
    float bv00[8], bv01[8], bv10[8], bv11[8];
    int   bi00[8], bi01[8], bi10[8], bi11[8];
#pragma unroll
    for (int i = 0; i < 8; ++i) {
        bv00[i] = bv01[i] = bv10[i] = bv11[i] = 3.4e38f;
        bi00[i] = bi01[i] = bi10[i] = bi11[i] = 0;
    }

    const int ntiles = Rpad / 16;
    const int nt2    = ntiles & ~1;

    // prologue: load tiles 0 and 1
    float2 rb0 = bt[(size_t)(col)      * 2 + hi];
    float2 rb1 = bt[(size_t)(16 + col) * 2 + hi];

    for (int t = 0; t < nt2; t += 2) {
        // look-ahead loads for tiles t+2, t+3 (always in-bounds via TPAD)
        const int nn0 = (t + 2) * 16 + col;
        float2 nb0 = bt[(size_t)nn0 * 2 + hi];
        float2 nb1 = bt[(size_t)(nn0 + 16) * 2 + hi];
        __builtin_prefetch(&bt[(size_t)(nn0 + 96) * 2 + hi], 0, 1);

        v2f B0; B0[0] = rb0.x; B0[1] = rb0.y;
        v2f B1; B1[0] = rb1.x; B1[1] = rb1.y;
        const int n0 = t * 16 + col;
        const int n1 = n0 + 16;

        v8f c0 = wmma4(a0, B0);
        v8f c1 = wmma4(a1, B0);
        v8f c2 = wmma4(a0, B1);
        v8f c3 = wmma4(a1, B1);
        UPDATE(bv00, bi00, c0, n0);
        UPDATE(bv10, bi10, c1, n0);
        UPDATE(bv01, bi01, c2, n1);
        UPDATE(bv11, bi11, c3, n1);

        rb0 = nb0; rb1 = nb1;
    }
    if (ntiles & 1) {
        v2f B0; B0[0] = rb0.x; B0[1] = rb0.y;
        const int n0 = nt2 * 16 + col;
        v8f c0 = wmma4(a0, B0);
        v8f c1 = wmma4(a1, B0);
        UPDATE(bv00, bi00, c0, n0);
        UPDATE(bv10, bi10, c1, n0);
    }

    // merge parity streams (tie -> smaller index preserves first-occurrence)
#pragma unroll
    for (int i = 0; i < 8; ++i) {
        if (bv01[i] < bv00[i] || (bv01[i] == bv00[i] && bi01[i] < bi00[i])) {
            bv00[i] = bv01[i]; bi00[i] = bi01[i];
        }
        if (bv11[i] < bv10[i] || (bv11[i] == bv10[i] && bi11[i] < bi10[i])) {
            bv10[i] = bv11[i]; bi10[i] = bi11[i];
        }
    }

    reduce_write(bv00, bi00, outIdx, (int)((size_t)b * P_) + qbase,      col, hi);
    reduce_write(bv10, bi10, outIdx, (int)((size_t)b * P_) + qbase + 16, col, hi);
}

// ---------------------------------------------------------------------------
// Accumulate face normals onto vertices (scatter with float atomics).
// ---------------------------------------------------------------------------
__global__ void face_normals(const float* __restrict__ verts,  // [B, N, 3]
                             const int*   __restrict__ faces,  // [B, F, 3]
                             float*       __restrict__ nrm) {  // [B, N, 3]
    int i = blockIdx.x * blockDim.x + threadIdx.x;
    if (i >= B_ * F_) return;
    int b = i / F_;
    int f = i - b * F_;
    const int* fc = faces + ((size_t)b * F_ + f) * 3;
    int i0 = fc[0], i1 = fc[1], i2 = fc[2];
    const float* vb = verts + (size_t)b * N_ * 3;
    float ax = vb[i0 * 3 + 0], ay = vb[i0 * 3 + 1], az = vb[i0 * 3 + 2];
    float bx = vb[i1 * 3 + 0], by = vb[i1 * 3 + 1], bz = vb[i1 * 3 + 2];
    float cx = vb[i2 * 3 + 0], cy = vb[i2 * 3 + 1], cz = vb[i2 * 3 + 2];
    float e1x = bx - ax, e1y = by - ay, e1z = bz - az;
    float e2x = cx - ax, e2y = cy - ay, e2z = cz - az;
    float nx = e1y * e2z - e1z * e2y;
    float ny = e1z * e2x - e1x * e2z;
    float nz = e1x * e2y - e1y * e2x;
    float* nb = nrm + (size_t)b * N_ * 3;
    atomicAdd(&nb[i0 * 3 + 0], nx); atomicAdd(&nb[i0 * 3 + 1], ny); atomicAdd(&nb[i0 * 3 + 2], nz);
    atomicAdd(&nb[i1 * 3 + 0], nx); atomicAdd(&nb[i1 * 3 + 1], ny); atomicAdd(&nb[i1 * 3 + 2], nz);
    atomicAdd(&nb[i2 * 3 + 0], nx); atomicAdd(&nb[i2 * 3 + 1], ny); atomicAdd(&nb[i2 * 3 + 2], nz);
}

// ---------------------------------------------------------------------------
// Gather, normalize, |dot|, mean-reduce to scalar.
// ---------------------------------------------------------------------------
__global__ __launch_bounds__(256) void final_loss(
    const float* __restrict__ pp,    // pred_points   [B, P, 3]
    const float* __restrict__ pv,    // pred_vertices [B, V, 3]
    const float* __restrict__ nrm,   // gt normals    [B, N, 3]
    const int*   __restrict__ nnGt,  // [B, P]
    const int*   __restrict__ nnPv,  // [B, P]
    float*       __restrict__ out) {
    __shared__ float sdata[256];
    int i = blockIdx.x * blockDim.x + threadIdx.x;
    float contrib = 0.0f;
    if (i < B_ * P_) {
        int b = i / P_;
        const float* q = pp + (size_t)i * 3;
        int jv = nnPv[i];
        const float* w = pv + ((size_t)b * V_ + jv) * 3;
        float ex = q[0] - w[0], ey = q[1] - w[1], ez = q[2] - w[2];
        float el = sqrtf(ex * ex + ey * ey + ez * ez);
        float ei = 1.0f / fmaxf(el, EPSF);
        ex *= ei; ey *= ei; ez *= ei;
        int jn = nnGt[i];
        const float* g = nrm + ((size_t)b * N_ + jn) * 3;
        float gx = g[0], gy = g[1], gz = g[2];
        float gl = sqrtf(gx * gx + gy * gy + gz * gz);
        float gi = 1.0f / fmaxf(gl, EPSF);
        gx *= gi; gy *= gi; gz *= gi;
        contrib = fabsf(ex * gx + ey * gy + ez * gz) * (1.0f / (float)(B_ * P_));
    }
    sdata[threadIdx.x] = contrib;
    __syncthreads();
    for (int s = 128; s > 0; s >>= 1) {
        if (threadIdx.x < s) sdata[threadIdx.x] += sdata[threadIdx.x + s];
        __syncthreads();
    }
    if (threadIdx.x == 0) atomicAdd(out, sdata[0]);
}

// ---------------------------------------------------------------------------
extern "C" void kernel_launch(void* const* d_in, const int* in_sizes, int n_in,
                              void* d_out, int out_size, void* d_ws, size_t ws_size,
                              hipStream_t stream) {
    (void)in_sizes; (void)n_in; (void)out_size; (void)ws_size;
    const float* pred_points = (const float*)d_in[0];
    const float* pred_verts  = (const float*)d_in[1];
    const float* gt_verts    = (const float*)d_in[2];
    const int*   gt_faces    = (const int*)d_in[3];

    char* ws = (char*)d_ws;
    float4* tabGt = (float4*)ws;  ws += (size_t)B_ * (N_   + TPAD) * sizeof(float4);
    float4* tabPv = (float4*)ws;  ws += (size_t)B_ * (VPAD + TPAD) * sizeof(float4);
    int*    nnGt  = (int*)ws;     ws += (size_t)B_ * P_ * sizeof(int);
    int*    nnPv  = (int*)ws;     ws += (size_t)B_ * P_ * sizeof(int);
    float*  nrm   = (float*)ws;   ws += (size_t)B_ * N_ * 3 * sizeof(float);

    hipMemsetAsync(nrm, 0, (size_t)B_ * N_ * 3 * sizeof(float), stream);
    hipMemsetAsync(d_out, 0, sizeof(float), stream);

    { int tot = B_ * (N_ + TPAD);
      build_reftab<<<(tot + 255) / 256, 256, 0, stream>>>(gt_verts, tabGt, N_, N_ + TPAD); }
    { int tot = B_ * (VPAD + TPAD);
      build_reftab<<<(tot + 255) / 256, 256, 0, stream>>>(pred_verts, tabPv, V_, VPAD + TPAD); }

    nn_argmin_wmma<<<B_ * (P_ / 32), 32, 0, stream>>>(pred_points, tabGt, nnGt, N_);
    nn_argmin_wmma<<<B_ * (P_ / 32), 32, 0, stream>>>(pred_points, tabPv, nnPv, VPAD);

    face_normals<<<(B_ * F_ + 255) / 256, 256, 0, stream>>>(gt_verts, gt_faces, nrm);

    final_loss<<<(B_ * P_ + 255) / 256, 256, 0, stream>>>(
        pred_points, pred_verts, nrm, nnGt, nnPv, (float*)d_out);
}